// GraphEncoder_53618371723609
// MI455X (gfx1250) — compile-verified
//
#include <hip/hip_runtime.h>
#include <hip/hip_bf16.h>

// Problem constants (match reference)
#define Bq 32
#define Nq 2048
#define Eq 16384
#define Dq 256
#define Rq 8
#define Lq 2
#define LN_EPS 1e-5f

typedef float v2f __attribute__((ext_vector_type(2)));
typedef float v8f __attribute__((ext_vector_type(8)));

// ------------------------------------------------------------------
// 1) Embedding gather: x[b,n,:] = emb[nodes[b,n],:]
// ------------------------------------------------------------------
__global__ void rgcn_embed(const int* __restrict__ nodes,
                           const float* __restrict__ emb,
                           float* __restrict__ x) {
    size_t idx = (size_t)blockIdx.x * blockDim.x + threadIdx.x;  // B*N*(D/4)
    int    dv  = (int)(idx & 63);        // D/4 = 64
    size_t bn  = idx >> 6;
    int node   = nodes[bn];
    const float4* s = (const float4*)(emb + (size_t)node * Dq);
    float4*       d = (float4*)(x + bn * Dq);
    d[dv] = s[dv];
}

// ------------------------------------------------------------------
// 2) Edge scatter: S[g,r,dst,:] += x[b,src,:], cnt[g,r,dst] += 1
//    64 lanes per edge, float4 per lane. f32 global atomics.
// ------------------------------------------------------------------
__global__ void rgcn_scatter(const int* __restrict__ edges,
                             const int* __restrict__ types,
                             const float* __restrict__ x,
                             float* __restrict__ S,
                             float* __restrict__ cnt,
                             int b0) {
    size_t idx = (size_t)blockIdx.x * blockDim.x + threadIdx.x;  // g*E*64
    int    c   = (int)(idx & 63);
    size_t t   = idx >> 6;
    int    e   = (int)(t & (Eq - 1));
    int    g   = (int)(t >> 14);         // /E, E = 16384
    int    b   = b0 + g;
    int sN = edges[((size_t)b * 2 + 0) * Eq + e];
    int dN = edges[((size_t)b * 2 + 1) * Eq + e];
    int r  = types[(size_t)b * Eq + e];
    float4 v = *(const float4*)(x + ((size_t)b * Nq + sN) * Dq + c * 4);
    size_t seg = ((size_t)g * Rq + r) * Nq + dN;
    float* sp = S + seg * Dq + c * 4;
    atomicAdd(sp + 0, v.x);
    atomicAdd(sp + 1, v.y);
    atomicAdd(sp + 2, v.z);
    atomicAdd(sp + 3, v.w);
    if (c == 0) atomicAdd(cnt + seg, 1.0f);
}

// ------------------------------------------------------------------
// 3) Fused RGCN GEMM + bias + LayerNorm + residual + ReLU.
//    Block = 256 threads (8 wave32), computes 64 rows x 256 cols.
//    Wave w: rows [n0 + (w>>1)*16, +16), cols [(w&1)*128, +128)
//    K loop: 9 passes (8 relations with 1/cnt scaling, then root on x).
//    Uses V_WMMA_F32_16X16X4_F32.
// ------------------------------------------------------------------
__global__ __launch_bounds__(256)
void rgcn_gemm_ln(const float* __restrict__ x,
                  const float* __restrict__ S,
                  const float* __restrict__ cnt,
                  const float* __restrict__ Wl,     // [R,D,D]
                  const float* __restrict__ rootl,  // [D,D]
                  const float* __restrict__ biasl,  // [D]
                  const float* __restrict__ gamma,
                  const float* __restrict__ beta,
                  float* __restrict__ out,          // [B,N,D]
                  int b0) {
    constexpr int STR = Dq + 1;                 // LDS pad to dodge bank conflicts
    __shared__ float tile[32 * STR];            // 32-row staging (two passes)
    __shared__ float mu_s[32];
    __shared__ float rs_s[32];

    const int g    = blockIdx.y;
    const int b    = b0 + g;
    const int n0   = blockIdx.x * 64;
    const int tid  = threadIdx.x;
    const int lane = tid & 31;
    const int w    = tid >> 5;
    const int wrow = w >> 1;                    // 0..3
    const int wcol = w & 1;                     // 0..1
    const int lm   = lane & 15;
    const int lh   = lane >> 4;                 // 0/1
    const int arow = n0 + wrow * 16 + lm;       // A-matrix row for this lane
    const int kh   = lh * 2;                    // K offset within 4-wide step

    v8f acc[8] = {};                            // 8 C tiles of 16x16 f32

    for (int r = 0; r < 9; ++r) {
        const float* Ap;
        const float* Bp;
        float sc;
        if (r < 8) {
            size_t seg = ((size_t)g * Rq + r) * Nq + arow;
            Ap = S + seg * Dq;
            sc = 1.0f / fmaxf(cnt[seg], 1.0f);  // mean scaling (max(cnt,1))
            Bp = Wl + (size_t)r * Dq * Dq;
        } else {
            Ap = x + ((size_t)b * Nq + arow) * Dq;
            sc = 1.0f;
            Bp = rootl;
        }
        #pragma unroll 4
        for (int k4 = 0; k4 < Dq; k4 += 4) {
            v2f a = *(const v2f*)(Ap + k4 + kh);   // K=k4+kh, k4+kh+1 (ISA A 16x4 layout)
            a *= sc;
            #pragma unroll
            for (int t = 0; t < 8; ++t) {
                int bn = wcol * 128 + t * 16 + lm; // B col (N) for this lane
                v2f bf;
                bf.x = Bp[(size_t)(k4 + kh) * Dq + bn];
                bf.y = Bp[(size_t)(k4 + kh + 1) * Dq + bn];
                acc[t] = __builtin_amdgcn_wmma_f32_16x16x4_f32(
                    false, a, false, bf, (short)0, acc[t], false, false);
            }
        }
    }

    // ---- Epilogue: bias + LayerNorm + residual + ReLU, 32 rows per pass ----
    const float gcol = gamma[tid];   // tid == output column in write phase
    const float bcol = beta[tid];
    #pragma unroll
    for (int p = 0; p < 2; ++p) {
        if ((wrow >> 1) == p) {      // waves owning rows [p*32, p*32+32)
            int rbase = (wrow & 1) * 16;
            #pragma unroll
            for (int t = 0; t < 8; ++t) {
                int col = wcol * 128 + t * 16 + lm;
                float bb = biasl[col];
                #pragma unroll
                for (int v = 0; v < 8; ++v) {
                    int row = rbase + v + lh * 8;  // C/D layout: VGPR v -> M=v / v+8
                    tile[row * STR + col] = acc[t][v] + bb;
                }
            }
        }
        __syncthreads();
        if (tid < 32) {
            float sum = 0.0f;
            #pragma unroll 4
            for (int c2 = 0; c2 < Dq; ++c2) sum += tile[tid * STR + c2];
            float mu = sum * (1.0f / Dq);
            float vs = 0.0f;
            #pragma unroll 4
            for (int c2 = 0; c2 < Dq; ++c2) {
                float d = tile[tid * STR + c2] - mu;
                vs += d * d;
            }
            mu_s[tid] = mu;
            rs_s[tid] = rsqrtf(vs * (1.0f / Dq) + LN_EPS);
        }
        __syncthreads();
        for (int rr = 0; rr < 32; ++rr) {
            int row = p * 32 + rr;
            float h = (tile[rr * STR + tid] - mu_s[rr]) * rs_s[rr] * gcol + bcol;
            size_t off = ((size_t)b * Nq + n0 + row) * Dq + tid;
            float o = h + x[off];                 // residual (read-then-write, in place safe)
            out[off] = fmaxf(o, 0.0f);
        }
        __syncthreads();
    }
}

// ------------------------------------------------------------------
// Host driver: embed, then per layer: (per graph-chunk) zero S/cnt,
// scatter, fused GEMM+LN. Chunk size adapts to ws_size.
// ------------------------------------------------------------------
extern "C" void kernel_launch(void* const* d_in, const int* in_sizes, int n_in,
                              void* d_out, int out_size, void* d_ws, size_t ws_size,
                              hipStream_t stream) {
    (void)in_sizes; (void)n_in; (void)out_size;
    const int*   nodes = (const int*)d_in[0];
    const int*   edges = (const int*)d_in[1];
    const int*   types = (const int*)d_in[2];
    const float* emb   = (const float*)d_in[3];
    const float* W     = (const float*)d_in[4];
    const float* root  = (const float*)d_in[5];
    const float* bias  = (const float*)d_in[6];
    const float* gamma = (const float*)d_in[7];
    const float* beta  = (const float*)d_in[8];
    float*       out   = (float*)d_out;

    float* x = (float*)d_ws;                       // [B,N,D] activations
    size_t xElems = (size_t)Bq * Nq * Dq;
    size_t xBytes = xElems * sizeof(float);
    size_t perG   = ((size_t)Rq * Nq * Dq + (size_t)Rq * Nq) * sizeof(float);
    size_t avail  = (ws_size > xBytes) ? (ws_size - xBytes) : 0;
    int G = (int)(avail / perG);
    if (G > Bq) G = Bq;
    if (G < 1)  G = 1;                             // best effort
    float* S   = (float*)((char*)d_ws + xBytes);   // [G,R,N,D]
    float* cnt = S + (size_t)G * Rq * Nq * Dq;     // [G,R,N]

    // 1) embedding gather
    rgcn_embed<<<(unsigned)(xElems / 4 / 256), 256, 0, stream>>>(nodes, emb, x);

    // 2) layers
    for (int l = 0; l < Lq; ++l) {
        const float* Wl    = W    + (size_t)l * Rq * Dq * Dq;
        const float* rootl = root + (size_t)l * Dq * Dq;
        const float* biasl = bias + (size_t)l * Dq;
        float* dst = (l == Lq - 1) ? out : x;
        for (int b0 = 0; b0 < Bq; b0 += G) {
            int g = (Bq - b0 < G) ? (Bq - b0) : G;
            hipMemsetAsync(S,   0, (size_t)g * Rq * Nq * Dq * sizeof(float), stream);
            hipMemsetAsync(cnt, 0, (size_t)g * Rq * Nq * sizeof(float), stream);
            size_t sthreads = (size_t)g * Eq * 64;
            rgcn_scatter<<<(unsigned)(sthreads / 256), 256, 0, stream>>>(
                edges, types, x, S, cnt, b0);
            dim3 grid(Nq / 64, (unsigned)g);
            rgcn_gemm_ln<<<grid, 256, 0, stream>>>(
                x, S, cnt, Wl, rootl, biasl, gamma, beta, dst, b0);
        }
    }
}